// KnnMSE_3642132267673
// MI455X (gfx1250) — compile-verified
//
#include <hip/hip_runtime.h>
#include <hip/hip_bf16.h>

typedef float v2f __attribute__((ext_vector_type(2)));
typedef float v8f __attribute__((ext_vector_type(8)));

#define ROWW 131      // 3 coords + 128 features per row
#define DF   128
#define KNN  8
#define NBAT 64
#define BLOCK 256
#define WAVES 8       // 256 threads / wave32
#define EPSV 1e-16f
#define INFV __builtin_inff()

// ---------------------------------------------------------------------------
// Kernel 1: batch boundaries of sorted b1.  starts[b] = lower_bound(b1, b),
// ends[b] = lower_bound(b1, b+1).  64 threads, one per batch id.
// ---------------------------------------------------------------------------
__global__ void knn_bounds_kernel(const int* __restrict__ b1, int n,
                                  int* __restrict__ starts, int* __restrict__ ends) {
    int b = threadIdx.x;
    if (b >= NBAT) return;
    int lo = 0, hi = n;
    while (lo < hi) { int mid = (lo + hi) >> 1; if (b1[mid] < b) lo = mid + 1; else hi = mid; }
    starts[b] = lo;
    lo = 0; hi = n;
    while (lo < hi) { int mid = (lo + hi) >> 1; if (b1[mid] < b + 1) lo = mid + 1; else hi = mid; }
    ends[b] = lo;
}

// ---------------------------------------------------------------------------
// Kernel 2: fused KNN + inverse-distance interpolation + squared error.
// One wave32 handles 32 consecutive queries; pairwise dot products via
// V_WMMA_F32_16X16X4_F32 (K=3 zero-padded to 4).
// ---------------------------------------------------------------------------
__global__ __launch_bounds__(BLOCK) void knn_mse_kernel(
    const float* __restrict__ x1, const float* __restrict__ x2,
    const int*  __restrict__ b1, const int*  __restrict__ b2,
    const int*  __restrict__ starts, const int* __restrict__ ends,
    float* __restrict__ partials, int n)
{
    __shared__ float sDot[WAVES][32][17];   // dot-product tile, padded (17 coprime w/ 64 banks)
    __shared__ float sN1 [WAVES][16];       // candidate squared norms
    __shared__ int   sB1 [WAVES][16];       // candidate batch ids
    __shared__ float sRed[BLOCK];

    const int lane = threadIdx.x & 31;
    const int wave = threadIdx.x >> 5;
    const int i0   = (blockIdx.x * WAVES + wave) * 32;   // wave's first query
    const int i    = i0 + lane;                          // this lane's query
    const bool iv  = (i < n);
    const int  iq  = iv ? i : (n - 1);

    // ---- per-lane query data (row = this lane's query) ----
    const float* q = x2 + (size_t)iq * ROWW;
    const float qx = q[0], qy = q[1], qz = q[2];
    const float n2i = qx * qx + qy * qy + qz * qz;
    const int   myB = b2[iq];

    // ---- build A tiles per ISA 16x4 f32 layout:
    // lanes 0-15: row M=lane, v0=K0(x), v1=K1(y); lanes 16-31: row M=lane-16, v0=K2(z), v1=K3(0)
    const int  mrow = lane & 15;
    const int  r0 = min(i0 + mrow,      n - 1);
    const int  r1 = min(i0 + 16 + mrow, n - 1);
    const float* qa0 = x2 + (size_t)r0 * ROWW;
    const float* qa1 = x2 + (size_t)r1 * ROWW;
    const bool lohalf = (lane < 16);
    v2f A0, A1;
    A0[0] = lohalf ? qa0[0] : qa0[2];
    A0[1] = lohalf ? qa0[1] : 0.0f;
    A1[0] = lohalf ? qa1[0] : qa1[2];
    A1[1] = lohalf ? qa1[1] : 0.0f;

    // ---- wave-uniform candidate range (b2 sorted => union range is contiguous) ----
    const int candStart = starts[b2[min(i0, n - 1)]];
    const int candEnd   = ends  [b2[min(i0 + 31, n - 1)]];

    // top-8 (ascending), register resident
    float bd[KNN]; int bi[KNN];
#pragma unroll
    for (int k = 0; k < KNN; ++k) { bd[k] = INFV; bi[k] = 0; }

    for (int j0 = candStart; j0 < candEnd; j0 += 16) {
        const int  c  = lane & 15;
        const int  j  = j0 + c;
        const bool jvld = (j < candEnd);
        const int  jc = jvld ? j : candStart;
        const float* p = x1 + (size_t)jc * ROWW;

        // B tile, 4x16 f32 (mirror of A layout); invalid cols zeroed + batch -1
        v2f Bm;
        if (lohalf) {
            float x = p[0], y = p[1], z = p[2];
            Bm[0] = jvld ? x : 0.0f;
            Bm[1] = jvld ? y : 0.0f;
            sN1[wave][c] = x * x + y * y + z * z;
            sB1[wave][c] = jvld ? b1[jc] : -1;
        } else {
            Bm[0] = jvld ? p[2] : 0.0f;
            Bm[1] = 0.0f;
        }

        // D = A x B : two 16x16 f32 tiles (rows i0..i0+15 and i0+16..i0+31)
        v8f d0 = {}; v8f d1 = {};
        d0 = __builtin_amdgcn_wmma_f32_16x16x4_f32(false, A0, false, Bm,
                                                   (short)0, d0, false, false);
        d1 = __builtin_amdgcn_wmma_f32_16x16x4_f32(false, A1, false, Bm,
                                                   (short)0, d1, false, false);

        // scatter D tiles to LDS per ISA C/D layout: vgpr r, lane l -> M = r + 8*(l>=16)
        const int rbase = (lane >> 4) * 8;
#pragma unroll
        for (int r = 0; r < 8; ++r) {
            sDot[wave][rbase + r][c]      = d0[r];
            sDot[wave][16 + rbase + r][c] = d1[r];
        }
        asm volatile("s_wait_dscnt 0x0" ::: "memory");   // wave-internal LDS RAW
        __builtin_amdgcn_wave_barrier();

        // scan: lane m owns query row m of the wave tile
#pragma unroll
        for (int cc = 0; cc < 16; ++cc) {
            float dot = sDot[wave][lane][cc];
            float d2  = fmaxf(n2i + sN1[wave][cc] - 2.0f * dot, 0.0f);
            d2 = (sB1[wave][cc] == myB) ? d2 : INFV;
            if (d2 < bd[KNN - 1]) {          // strict < keeps earlier index on ties (matches top_k)
                bd[KNN - 1] = d2; bi[KNN - 1] = j0 + cc;
#pragma unroll
                for (int s = KNN - 2; s >= 0; --s) {
                    if (bd[s + 1] < bd[s]) {
                        float tf = bd[s]; bd[s] = bd[s + 1]; bd[s + 1] = tf;
                        int   ti = bi[s]; bi[s] = bi[s + 1]; bi[s + 1] = ti;
                    }
                }
            }
        }
        asm volatile("s_wait_dscnt 0x0" ::: "memory");   // reads done before next tile overwrites
        __builtin_amdgcn_wave_barrier();
    }

    // ---- inverse-distance weights: w = 1/max(d2, EPS); 0 for missing (inf) ----
    float w[KNN]; float wsum = 0.0f;
#pragma unroll
    for (int k = 0; k < KNN; ++k) {
        float d2 = bd[k];
        float wk = (d2 < INFV) ? (1.0f / fmaxf(d2, EPSV)) : 0.0f;
        w[k] = wk; wsum += wk;
    }
    const float invw = 1.0f / wsum;

    // ---- gather + interpolate + squared error (all rows L2-resident) ----
    const float* fr[KNN];
#pragma unroll
    for (int k = 0; k < KNN; ++k) fr[k] = x1 + (size_t)bi[k] * ROWW + 3;
    const float* f2r = x2 + (size_t)iq * ROWW + 3;

    float err = 0.0f;
    for (int d = 0; d < DF; ++d) {
        float acc = 0.0f;
#pragma unroll
        for (int k = 0; k < KNN; ++k) acc += w[k] * fr[k][d];
        float diff = acc * invw - f2r[d];
        err += diff * diff;
    }
    if (!iv) err = 0.0f;

    // ---- deterministic block reduction -> per-block partial ----
    sRed[threadIdx.x] = err;
    __syncthreads();
    for (int s = BLOCK / 2; s > 0; s >>= 1) {
        if (threadIdx.x < s) sRed[threadIdx.x] += sRed[threadIdx.x + s];
        __syncthreads();
    }
    if (threadIdx.x == 0) partials[blockIdx.x] = sRed[0];
}

// ---------------------------------------------------------------------------
// Kernel 3: sum partials (fixed order -> deterministic), scale by 1/(N*D).
// ---------------------------------------------------------------------------
__global__ void knn_finalize_kernel(const float* __restrict__ partials,
                                    float* __restrict__ out, int nblocks, float scale) {
    __shared__ float s[64];
    int t = threadIdx.x;
    float acc = 0.0f;
    for (int idx = t; idx < nblocks; idx += 64) acc += partials[idx];
    s[t] = acc;
    __syncthreads();
    for (int st = 32; st > 0; st >>= 1) {
        if (t < st) s[t] += s[t + st];
        __syncthreads();
    }
    if (t == 0) out[0] = s[0] * scale;
}

// ---------------------------------------------------------------------------
extern "C" void kernel_launch(void* const* d_in, const int* in_sizes, int n_in,
                              void* d_out, int out_size, void* d_ws, size_t ws_size,
                              hipStream_t stream) {
    const float* x1 = (const float*)d_in[0];
    const float* x2 = (const float*)d_in[1];
    const int*   b1 = (const int*)d_in[2];
    const int*   b2 = (const int*)d_in[3];
    const int n = in_sizes[2];                 // N from b1 length

    int*   starts   = (int*)d_ws;              // [64]
    int*   ends     = starts + NBAT;           // [64]
    float* partials = (float*)(starts + 2 * NBAT);

    knn_bounds_kernel<<<1, NBAT, 0, stream>>>(b1, n, starts, ends);

    const int nblocks = (n + BLOCK - 1) / BLOCK;   // 256 queries per block
    knn_mse_kernel<<<nblocks, BLOCK, 0, stream>>>(x1, x2, b1, b2, starts, ends,
                                                  partials, n);

    knn_finalize_kernel<<<1, 64, 0, stream>>>(partials, (float*)d_out, nblocks,
                                              1.0f / ((float)n * (float)DF));
}